// TripletLossCustom_16862041604543
// MI455X (gfx1250) — compile-verified
//
#include <hip/hip_runtime.h>
#include <math.h>
#include <stdint.h>

typedef __attribute__((ext_vector_type(2))) float v2f;
typedef __attribute__((ext_vector_type(4))) float v4f;
typedef __attribute__((ext_vector_type(8))) float v8f;

#define TRIP_MARGIN 1.0f
#define TRIP_EPS 1e-12f

constexpr int D_DIM = 512;
constexpr int N_NEG = 16;
constexpr int WAVES_PER_BLOCK = 8;
constexpr int BLOCK_THREADS = WAVES_PER_BLOCK * 32;

constexpr int CHUNK_COLS = 128;                    // columns per async-staged chunk
constexpr int N_CHUNKS = D_DIM / CHUNK_COLS;       // 4
constexpr int ROW_STRIDE_F = CHUNK_COLS + 4;       // 132 floats: pad -> conflict-free ds reads
constexpr int TILE_FLOATS = N_NEG * ROW_STRIDE_F;  // 2112 floats per buffer
constexpr int ANCHOR_FLOATS = WAVES_PER_BLOCK * D_DIM;
constexpr size_t SMEM_FLOATS =
    (size_t)ANCHOR_FLOATS + (size_t)WAVES_PER_BLOCK * 2 * TILE_FLOATS + WAVES_PER_BLOCK;
constexpr size_t SMEM_BYTES = SMEM_FLOATS * 4;     // ~151.6 KB (within 320 KB WGP LDS)

// One wave (32 lanes) per batch element b.
// negs sum-of-squares via fp32 WMMA Gram trick: C += A*A, diag(C) = ||diff_n||^2.
// negs tile streamed global->LDS with async copies (ASYNCcnt), double buffered.
__global__ __launch_bounds__(BLOCK_THREADS) void triplet_wmma_kernel(
    const float* __restrict__ anchor,
    const float* __restrict__ pos,
    const float* __restrict__ negs,
    float* __restrict__ blockPartials,
    int B)
{
    extern __shared__ float smem[];
    float* sA = smem + (size_t)(threadIdx.x >> 5) * D_DIM;                   // this wave's anchor
    float* sN = smem + ANCHOR_FLOATS + (size_t)(threadIdx.x >> 5) * (2 * TILE_FLOATS);
    float* sW = smem + ANCHOR_FLOATS + (size_t)WAVES_PER_BLOCK * 2 * TILE_FLOATS;

    const int lane = threadIdx.x & 31;
    const int wave = threadIdx.x >> 5;
    const int b = blockIdx.x * WAVES_PER_BLOCK + wave;

    float result = 0.0f;

    if (b < B) {
        const float* anc = anchor + (size_t)b * D_DIM;
        const float* ps  = pos    + (size_t)b * D_DIM;
        const char*  negBase = (const char*)(negs + (size_t)b * N_NEG * D_DIM);

        // ---- kick off async copy of chunk 0 (overlaps the d_pos phase) ----
        // chunk c, buffer p: 16 rows, one fully-coalesced b128 instruction per row.
        #pragma unroll
        for (int it = 0; it < N_NEG; ++it) {
            unsigned laddr = (unsigned)(uintptr_t)(sN) + it * (ROW_STRIDE_F * 4) + lane * 16;
            uint64_t gaddr = (uint64_t)(uintptr_t)negBase + (size_t)it * (D_DIM * 4) + lane * 16;
            asm volatile("global_load_async_to_lds_b128 %0, %1, off"
                         :: "v"(laddr), "v"(gaddr) : "memory");
        }

        // ---- stage anchor into LDS (coalesced b128), accumulate d_pos partial ----
        float dp = 0.0f;
        #pragma unroll
        for (int it = 0; it < D_DIM / (32 * 4); ++it) {   // 4 iterations
            const int idx4 = (it * 32 + lane) * 4;
            v4f a4 = *(const v4f*)(anc + idx4);
            v4f p4 = *(const v4f*)(ps  + idx4);
            *(v4f*)&sA[idx4] = a4;
            v4f dv = a4 - p4;
            dp += dv.x * dv.x + dv.y * dv.y + dv.z * dv.z + dv.w * dv.w;
        }
        asm volatile("s_wait_dscnt 0" ::: "memory");      // same-wave LDS store->load

        #pragma unroll
        for (int off = 16; off > 0; off >>= 1)
            dp += __shfl_xor(dp, off, 32);
        const float d_pos = sqrtf(dp + TRIP_EPS);

        // ---- negatives: double-buffered async-LDS staging feeding f32 WMMA ----
        // A-operand (16x4 f32): lanes 0-15 -> M=lane, K{0,1}; lanes 16-31 -> M=lane-16, K{2,3}.
        const int row  = lane & 15;
        const int koff = (lane >> 4) * 2;

        v8f acc = {};
        for (int c = 0; c < N_CHUNKS; ++c) {
            if (c + 1 < N_CHUNKS) {
                // issue next chunk into the other buffer
                const int p1 = (c + 1) & 1;
                unsigned lbase = (unsigned)(uintptr_t)(sN + p1 * TILE_FLOATS);
                uint64_t gbase = (uint64_t)(uintptr_t)negBase + (size_t)(c + 1) * (CHUNK_COLS * 4);
                #pragma unroll
                for (int it = 0; it < N_NEG; ++it) {
                    unsigned laddr = lbase + it * (ROW_STRIDE_F * 4) + lane * 16;
                    uint64_t gaddr = gbase + (size_t)it * (D_DIM * 4) + lane * 16;
                    asm volatile("global_load_async_to_lds_b128 %0, %1, off"
                                 :: "v"(laddr), "v"(gaddr) : "memory");
                }
                // async loads complete in order: <=16 outstanding => chunk c is resident
                asm volatile("s_wait_asynccnt 0x10" ::: "memory");
            } else {
                asm volatile("s_wait_asynccnt 0x0" ::: "memory");
            }

            const float* tile = sN + (c & 1) * TILE_FLOATS;
            const float* nRow = tile + (size_t)row * ROW_STRIDE_F + koff;  // conflict-free banks
            const float* aCol = sA + c * CHUNK_COLS + koff;                // broadcast reads

            #pragma unroll 8
            for (int d0 = 0; d0 < CHUNK_COLS; d0 += 4) {
                v2f nv = *(const v2f*)(nRow + d0);
                v2f av = *(const v2f*)(aCol + d0);
                v2f df = av - nv;
                acc = __builtin_amdgcn_wmma_f32_16x16x4_f32(
                    false, df, false, df, (short)0, acc, false, false);
            }
        }

        // ---- diag(C): n<8 -> VGPR n lane n ; n>=8 -> VGPR n-8 lane n+16 ----
        float diag = 0.0f;
        #pragma unroll
        for (int i = 0; i < 8; ++i) {
            diag = (lane == i)      ? acc[i] : diag;
            diag = (lane == 24 + i) ? acc[i] : diag;
        }
        const bool hasDiag = (lane < 8) || (lane >= 24);
        float dn = hasDiag ? sqrtf(diag + TRIP_EPS) : 0.0f;
        #pragma unroll
        for (int off = 16; off > 0; off >>= 1)
            dn += __shfl_xor(dn, off, 32);

        const float per = (float)N_NEG * d_pos - dn + TRIP_MARGIN;
        result = per > 0.0f ? per : 0.0f;
    }

    if (lane == 0) sW[wave] = result;
    __syncthreads();
    if (threadIdx.x == 0) {
        float s = 0.0f;
        #pragma unroll
        for (int w = 0; w < WAVES_PER_BLOCK; ++w) s += sW[w];
        blockPartials[blockIdx.x] = s;
    }
}

// Deterministic final reduction of per-block partials (single block).
__global__ __launch_bounds__(256) void triplet_finalize_kernel(
    const float* __restrict__ partials, int n, float* __restrict__ out)
{
    __shared__ float sdata[256];
    float s = 0.0f;
    for (int i = threadIdx.x; i < n; i += 256) s += partials[i];
    sdata[threadIdx.x] = s;
    __syncthreads();
    #pragma unroll
    for (int stride = 128; stride > 0; stride >>= 1) {
        if ((int)threadIdx.x < stride) sdata[threadIdx.x] += sdata[threadIdx.x + stride];
        __syncthreads();
    }
    if (threadIdx.x == 0) out[0] = sdata[0];
}

extern "C" void kernel_launch(void* const* d_in, const int* in_sizes, int n_in,
                              void* d_out, int out_size, void* d_ws, size_t ws_size,
                              hipStream_t stream) {
    const float* anchor = (const float*)d_in[0];   // [B, 512]
    const float* pos    = (const float*)d_in[1];   // [B, 512]
    const float* negs   = (const float*)d_in[2];   // [B, 16, 512]
    float* out = (float*)d_out;

    const int B = in_sizes[0] / D_DIM;
    const int blocks = (B + WAVES_PER_BLOCK - 1) / WAVES_PER_BLOCK;
    float* partials = (float*)d_ws;

    triplet_wmma_kernel<<<blocks, BLOCK_THREADS, SMEM_BYTES, stream>>>(
        anchor, pos, negs, partials, B);
    triplet_finalize_kernel<<<1, 256, 0, stream>>>(partials, blocks, out);
}